// SpikingNeuron_78142634983442
// MI455X (gfx1250) — compile-verified
//
#include <hip/hip_runtime.h>

// LIF scan, MI455X (gfx1250).
// Roofline: 4004 x 32768 x 4B = 524.8 MB of one-shot streaming output
// -> ~22.5 us floor @ 23.3 TB/s HBM; compute is ~0.2 G lane-ops, negligible.
// No matrix math exists in this op, so no WMMA: the CDNA5 levers here are
// wave32 occupancy, 128-bit non-temporal stores (stream > 192 MB L2, written
// once, never re-read), global_prefetch, and (element-tile x time-chunk)
// parallelism where each block bit-exactly fast-forwards the recurrence from
// t=0 (fma+cmp+cndmask per step) before emitting its time slice.

namespace {
constexpr int   kSteps      = 2001;
constexpr int   kElems      = 16 * 2048;            // 32768
constexpr float kLeak       = 0.95f;                // f32(0.95) — same either way
// Reference computes (1.0 - LEAK) in double, rounded to f32 at use:
// f32(0.04999999999999999) = 0x3d4ccccd, NOT 1.0f-0.95f = 0x3d4cccd0.
constexpr float kDrive      = (float)(1.0 - 0.95);
constexpr float kThr        = 0.3f;                 // f32(0.3) = 0x3e99999a
constexpr int   kThreads    = 256;                  // 8 wave32 per block
constexpr int   kVec        = 4;                    // float4 per lane -> b128 stores
constexpr int   kElemBlocks = kElems / (kThreads * kVec); // 32
constexpr int   kChunks     = 16;                   // recompute factor ~8.5x (free vs BW)
constexpr int   kChunkSteps = (kSteps + kChunks - 1) / kChunks; // 126
}

typedef float v4f __attribute__((ext_vector_type(4)));

__global__ __launch_bounds__(kThreads)
void lif_scan_kernel(const float* __restrict__ x, float* __restrict__ out) {
    const int e  = (blockIdx.x * kThreads + threadIdx.x) * kVec;
    const int t0 = blockIdx.y * kChunkSteps;
    int t1 = t0 + kChunkSteps;
    if (t1 > kSteps) t1 = kSteps;

    // Output layout (flat, reference return order):
    //   s_last[32768] | v_last[32768] | v_t[2001*32768] | s_t[2001*32768]
    float* __restrict__ s_last = out;
    float* __restrict__ v_last = out + kElems;
    float* __restrict__ v_t    = out + 2 * kElems;
    float* __restrict__ s_t    = v_t + (size_t)kSteps * kElems;

    __builtin_prefetch(x + e, 0, 3);
    const v4f xv = *(const v4f*)(x + e);

    v4f drive;
    #pragma unroll
    for (int i = 0; i < kVec; ++i) drive[i] = kDrive * xv[i];

    v4f v = {0.0f, 0.0f, 0.0f, 0.0f};

    // Fast-forward [0, t0): identical float op sequence as a serial run,
    // no stores. Worst block: ~1890 steps * 12 VALU ~ 9 us, hidden under the
    // 22.5 us store stream of the other chunks.
    for (int t = 0; t < t0; ++t) {
        #pragma unroll
        for (int i = 0; i < kVec; ++i) {
            const float nv = __builtin_fmaf(kLeak, v[i], drive[i]);
            v[i] = (nv > kThr) ? 0.0f : nv;
        }
    }

    // Emit slice [t0, t1): one v_t row + one s_t row per step, 128-bit NT
    // stores (wave writes 2 x 512 B contiguous per step).
    v4f sv = {0.0f, 0.0f, 0.0f, 0.0f};
    for (int t = t0; t < t1; ++t) {
        #pragma unroll
        for (int i = 0; i < kVec; ++i) {
            const float nv  = __builtin_fmaf(kLeak, v[i], drive[i]);
            const bool  spk = nv > kThr;
            sv[i] = spk ? 1.0f : 0.0f;   // int32 {0,1} in reference -> float lane
            v[i]  = spk ? 0.0f : nv;     // hard reset
        }
        const size_t row = (size_t)t * kElems + (size_t)e;
        __builtin_nontemporal_store(v,  (v4f*)(v_t + row));
        __builtin_nontemporal_store(sv, (v4f*)(s_t + row));
    }

    // Final chunk also owns s_last / v_last (values after step kSteps-1).
    if (t1 == kSteps) {
        __builtin_nontemporal_store(v,  (v4f*)(v_last + e));
        __builtin_nontemporal_store(sv, (v4f*)(s_last + e));
    }
}

extern "C" void kernel_launch(void* const* d_in, const int* in_sizes, int n_in,
                              void* d_out, int out_size, void* d_ws, size_t ws_size,
                              hipStream_t stream) {
    (void)in_sizes; (void)n_in; (void)out_size; (void)d_ws; (void)ws_size;
    const float* x = (const float*)d_in[0];
    float* out = (float*)d_out;
    dim3 grid(kElemBlocks, kChunks, 1);
    lif_scan_kernel<<<grid, dim3(kThreads, 1, 1), 0, stream>>>(x, out);
}